// Model_15418932593049
// MI455X (gfx1250) — compile-verified
//
#include <hip/hip_runtime.h>
#include <math.h>

// Attention-state merge (prefix/suffix flash-attention partials), MI455X.
// T=16384, H=32, D=128, fp32. Pure HBM-streaming kernel:
//   traffic ~810 MB, intensity ~0.33 FLOP/B -> memory floor ~35us @ 23.3 TB/s.
// Design: one wave32 per (t,h) row; b128 accesses (512 B per wave per vmem op);
// non-temporal TH hints for the 768 MB of single-use bulk traffic so the tiny,
// 32x-re-read LSE arrays stay resident in the 192 MB L2. Hardware
// transcendentals (v_exp/v_log/v_rcp) keep the per-row VALU count minimal —
// their arguments are range-safe here (exp arg <= 0, log arg in [1,2]).

typedef float float4v __attribute__((ext_vector_type(4)));

__global__ __launch_bounds__(256) void merge_attn_states_kernel(
    const float* __restrict__ p_out,    // [R, 128]  R = T*H
    const float* __restrict__ p_lse,    // [R]
    const float* __restrict__ s_out,    // [R, 128]
    const float* __restrict__ s_lse,    // [R]
    float* __restrict__ out,            // [R, 128]
    float* __restrict__ out_lse,        // [R]
    int n_rows)
{
    const int tid  = blockIdx.x * blockDim.x + threadIdx.x;
    const int row  = tid >> 5;          // one wave32 per (token, head) row
    const int lane = tid & 31;          // each lane covers 4 of 128 elements
    if (row >= n_rows) return;

    // Wave-uniform LSE loads -> single coalesced request, L2-resident (RT hint).
    float pl = p_lse[row];
    float sl = s_lse[row];
    // +inf is the "no state" sentinel -> treat as -inf (matches reference).
    if (pl == INFINITY) pl = -INFINITY;
    if (sl == INFINITY) sl = -INFINITY;

    const float m   = fmaxf(pl, sl);
    const float pe  = __expf(pl - m);          // v_exp_f32 (arg <= 0)
    const float se  = __expf(sl - m);          // v_exp_f32 (arg <= 0)
    const float sum = pe + se;                 // in (0, 2]
    const float inv = __builtin_amdgcn_rcpf(sum); // v_rcp_f32
    const float ps  = pe * inv;
    const float ss  = se * inv;

    const long long base = (long long)row * 128 + lane * 4;

    // Single-use bulk traffic: b128 + non-temporal hints (CDNA5 TH_*_NT).
    float4v pv = __builtin_nontemporal_load((const float4v*)(p_out + base));
    float4v sv = __builtin_nontemporal_load((const float4v*)(s_out + base));

    float4v ov;
    ov.x = fmaf(pv.x, ps, sv.x * ss);
    ov.y = fmaf(pv.y, ps, sv.y * ss);
    ov.z = fmaf(pv.z, ps, sv.z * ss);
    ov.w = fmaf(pv.w, ps, sv.w * ss);

    __builtin_nontemporal_store(ov, (float4v*)(out + base));

    if (lane == 0) {
        out_lse[row] = __logf(sum) + m;        // v_log_f32 (arg in (0,2])
    }
}

extern "C" void kernel_launch(void* const* d_in, const int* in_sizes, int n_in,
                              void* d_out, int out_size, void* d_ws, size_t ws_size,
                              hipStream_t stream) {
    (void)n_in; (void)d_ws; (void)ws_size; (void)out_size;

    const float* p_out = (const float*)d_in[0];  // prefix_output [T,H,D]
    const float* p_lse = (const float*)d_in[1];  // prefix_lse    [T,H]
    const float* s_out = (const float*)d_in[2];  // suffix_output [T,H,D]
    const float* s_lse = (const float*)d_in[3];  // suffix_lse    [T,H]

    const int n_rows = in_sizes[1];              // T*H = 524288 (D fixed at 128)

    float* out     = (float*)d_out;                // first T*H*D floats
    float* out_lse = out + (long long)in_sizes[0]; // next T*H floats

    // One wave32 per row -> 32 threads per row; 256-thread blocks = 8 waves.
    const int threads_per_block = 256;
    const long long total_threads = (long long)n_rows * 32;
    const int blocks = (int)((total_threads + threads_per_block - 1) / threads_per_block);

    merge_attn_states_kernel<<<blocks, threads_per_block, 0, stream>>>(
        p_out, p_lse, s_out, s_lse, out, out_lse, n_rows);
}